// HeteroSTBlock_52226802319859
// MI455X (gfx1250) — compile-verified
//
#include <hip/hip_runtime.h>
#include <math.h>

// ---------------- problem constants (match reference) ----------------
constexpr int kB   = 4;
constexpr int kT   = 64;
constexpr int kKt  = 3;
constexpr int kNr  = 256;
constexpr int kNd  = 1024;
constexpr int kNp  = 4096;
constexpr int kCin = 32;
constexpr int kCm  = 64;
constexpr int kCo  = 64;
constexpr int kT1  = kT - kKt + 1;   // 62
constexpr int kT2  = kT1 - kKt + 1;  // 60

typedef __attribute__((ext_vector_type(2))) float v2f;
typedef __attribute__((ext_vector_type(8))) float v8f;

// CDNA5 f32 WMMA: D(16x16) = A(16x4) * B(4x16) + C
__device__ __forceinline__ v8f wmma4(v2f a, v2f b, v8f c) {
  return __builtin_amdgcn_wmma_f32_16x16x4_f32(false, a, false, b, (short)0, c,
                                               false, false);
}
__device__ __forceinline__ float sigmoidf_(float x) {
  return 1.f / (1.f + __expf(-x));
}

// ---------------- stage 1: gated temporal conv (Cin=32, K=96) ----------------
// x: (B, Cin, T, N); W: (128, Cin, Kt); bias: (128); out: (B, Cm, T1, N)
// grid.x = M tiles; block = 128 (4 waves = 4 column tile pairs).
__global__ void tconv1_kernel(const float* __restrict__ x,
                              const float* __restrict__ W,
                              const float* __restrict__ bias,
                              float* __restrict__ out, int N, int M) {
  int lane = threadIdx.x & 31;
  int co0 = (threadIdx.x >> 5) * 16;  // lo cols; hi cols = co0+64
  int m0 = blockIdx.x * 16;
  int hf = lane >> 4, idx = lane & 15;

  int mA = m0 + idx;
  int mc = (mA < M) ? mA : (M - 1);  // clamp: bad rows never stored
  int nA = mc % N;
  int tA = (mc / N) % kT1;
  int bA = mc / (N * kT1);
  const float* xA = x + ((size_t)bA * kCin * kT + tA) * N + nA;
  const float* pWlo = W + (size_t)(co0 + idx) * (kCin * kKt) + 2 * hf;
  const float* pWhi = pWlo + (size_t)64 * (kCin * kKt);

  // incremental k -> (ci,kt) for the two k-slots (k = 4*ks + 2*hf + j)
  int k0 = 2 * hf, k1 = 2 * hf + 1;
  int kt0 = k0 % kKt, kt1 = k1 % kKt;
  int off0 = ((k0 / kKt) * kT + kt0) * N;
  int off1 = ((k1 / kKt) * kT + kt1) * N;
  const int dN = (kT + 1) * N;       // kt++, ci++
  const int dW = (2 * kT - 2) * N;   // kt 2->0, ci+=2

  v8f accL = {};
  v8f accH = {};
  for (int ks = 0; ks < (kCin * kKt) / 4; ++ks) {
    v2f a, bl, bh;
    a[0] = xA[off0];
    a[1] = xA[off1];
    bl[0] = pWlo[0]; bl[1] = pWlo[1];
    bh[0] = pWhi[0]; bh[1] = pWhi[1];
    accL = wmma4(a, bl, accL);
    accH = wmma4(a, bh, accH);
    if (kt0 < kKt - 1) { kt0++; off0 += dN; } else { kt0 = 0; off0 += dW; }
    if (kt1 < kKt - 1) { kt1++; off1 += dN; } else { kt1 = 0; off1 += dW; }
    pWlo += 4; pWhi += 4;
  }
#pragma unroll
  for (int r = 0; r < 8; ++r) {
    int m = m0 + r + 8 * hf;
    if (m >= M) continue;
    int n = m % N;
    int t = (m / N) % kT1;
    int b = m / (N * kT1);
    int co = co0 + idx;
    float ylo = accL[r] + bias[co];
    float yhi = accH[r] + bias[co + 64];
    float xin =
        (co < kCin) ? x[(((size_t)b * kCin + co) * kT + t + (kKt - 1)) * N + n]
                    : 0.f;
    out[(((size_t)b * kCm + co) * kT1 + t) * N + n] = (ylo + xin) * sigmoidf_(yhi);
  }
}

// ---------------- stage 2: gated temporal conv on relu(x) (Cin=64, K=192) ----
// LAYOUT 0: x is (B, Cm, T1, N).  LAYOUT 1: x is flat ((b*T1+t)*N+n, Cm).
template <int LAYOUT>
__device__ __forceinline__ float ld2(const float* x, int b, int c, int tt,
                                     int n, int N) {
  float v;
  if (LAYOUT == 0)
    v = x[(((size_t)b * kCm + c) * kT1 + tt) * N + n];
  else
    v = x[(((size_t)b * kT1 + tt) * N + n) * (size_t)kCm + c];
  return fmaxf(v, 0.f);
}

template <int LAYOUT>
__global__ void tconv2_kernel(const float* __restrict__ x,
                              const float* __restrict__ W,
                              const float* __restrict__ bias,
                              float* __restrict__ out, int N, int M) {
  int lane = threadIdx.x & 31;
  int co0 = (threadIdx.x >> 5) * 16;
  int m0 = blockIdx.x * 16;
  int hf = lane >> 4, idx = lane & 15;

  int mA = m0 + idx;
  int mc = (mA < M) ? mA : (M - 1);
  int nA = mc % N;
  int tA = (mc / N) % kT2;
  int bA = mc / (N * kT2);
  const float* xA;
  int off0, off1, dN, dW;
  int k0 = 2 * hf, k1 = 2 * hf + 1;
  int kt0 = k0 % kKt, kt1 = k1 % kKt;
  if (LAYOUT == 0) {
    xA = x + ((size_t)bA * kCm * kT1 + tA) * N + nA;
    off0 = ((k0 / kKt) * kT1 + kt0) * N;
    off1 = ((k1 / kKt) * kT1 + kt1) * N;
    dN = (kT1 + 1) * N;
    dW = (2 * kT1 - 2) * N;
  } else {
    xA = x + (((size_t)bA * kT1 + tA) * N + nA) * (size_t)kCm;
    off0 = kt0 * (N * kCm) + (k0 / kKt);
    off1 = kt1 * (N * kCm) + (k1 / kKt);
    dN = N * kCm + 1;
    dW = 2 - 2 * N * kCm;
  }
  const float* pWlo = W + (size_t)(co0 + idx) * (kCm * kKt) + 2 * hf;
  const float* pWhi = pWlo + (size_t)64 * (kCm * kKt);

  v8f accL = {};
  v8f accH = {};
  for (int ks = 0; ks < (kCm * kKt) / 4; ++ks) {
    v2f a, bl, bh;
    a[0] = fmaxf(xA[off0], 0.f);
    a[1] = fmaxf(xA[off1], 0.f);
    bl[0] = pWlo[0]; bl[1] = pWlo[1];
    bh[0] = pWhi[0]; bh[1] = pWhi[1];
    accL = wmma4(a, bl, accL);
    accH = wmma4(a, bh, accH);
    if (kt0 < kKt - 1) { kt0++; off0 += dN; } else { kt0 = 0; off0 += dW; }
    if (kt1 < kKt - 1) { kt1++; off1 += dN; } else { kt1 = 0; off1 += dW; }
    pWlo += 4; pWhi += 4;
  }
#pragma unroll
  for (int r = 0; r < 8; ++r) {
    int m = m0 + r + 8 * hf;
    if (m >= M) continue;
    int n = m % N;
    int t = (m / N) % kT2;
    int b = m / (N * kT2);
    int co = co0 + idx;
    float ylo = accL[r] + bias[co];
    float yhi = accH[r] + bias[co + 64];
    float xin = ld2<LAYOUT>(x, b, co, t + (kKt - 1), n, N);
    out[(((size_t)b * kCo + co) * kT2 + t) * N + n] = (ylo + xin) * sigmoidf_(yhi);
  }
}

// ---------------- LayerNorm over channels, in place on (B, Co, T2, N) -------
__global__ void ln_kernel(float* __restrict__ y, const float* __restrict__ g,
                          const float* __restrict__ bta, int N, int M) {
  int i = blockIdx.x * blockDim.x + threadIdx.x;
  if (i >= M) return;
  int n = i % N;
  int t = (i / N) % kT2;
  int b = i / (N * kT2);
  size_t base = (((size_t)b * kCo) * kT2 + t) * N + n;
  size_t cs = (size_t)kT2 * N;
  float mu = 0.f;
  for (int c = 0; c < kCo; ++c) mu += y[base + c * cs];
  mu *= (1.f / kCo);
  float var = 0.f;
  for (int c = 0; c < kCo; ++c) {
    float d = y[base + c * cs] - mu;
    var += d * d;
  }
  var *= (1.f / kCo);
  float rs = rsqrtf(var + 1e-5f);
  for (int c = 0; c < kCo; ++c)
    y[base + c * cs] = (y[base + c * cs] - mu) * rs * g[c] + bta[c];
}

// ---------------- misc small kernels ----------------------------------------
__global__ void fill_kernel(float* __restrict__ p, float v, size_t n) {
  size_t i = (size_t)blockIdx.x * blockDim.x + threadIdx.x;
  if (i < n) p[i] = v;
}

// segment-sum scatter for SAGE.  xs layout 0: tensor (B,Cm,T1,Ns); 1: flat rows
__global__ void sage_scatter_kernel(const float* __restrict__ xs, int layout,
                                    const int* __restrict__ ei0,
                                    const int* __restrict__ ei1, int E, int Ns,
                                    int Ndst, float* __restrict__ s,
                                    float* __restrict__ cnt) {
  int gid = blockIdx.x * blockDim.x + threadIdx.x;
  int total = kT1 * E * 16;
  if (gid >= total) return;
  int sub = gid & 15;
  int ee = gid >> 4;
  int e = ee % E;
  int t = ee / E;
  int srow = ei0[e] + t * kB * Ns;
  int drow = ei1[e] + t * kB * Ndst;
  const float* src;
  size_t stride;
  if (layout == 0) {
    int nn = srow % Ns;
    int tt = (srow / Ns) % kT1;
    int b = srow / (Ns * kT1);
    src = xs + (((size_t)b * kCm) * kT1 + tt) * Ns + nn;
    stride = (size_t)kT1 * Ns;
  } else {
    src = xs + (size_t)srow * kCm;
    stride = 1;
  }
  float* dsts = s + (size_t)drow * kCm;
#pragma unroll
  for (int j = 0; j < 4; ++j) {
    int c = sub * 4 + j;
    atomicAdd(&dsts[c], src[c * stride]);
  }
  if (sub == 0) atomicAdd(&cnt[drow], 1.f);
}

__global__ void mean_div_kernel(float* __restrict__ s,
                                const float* __restrict__ cnt, int rows) {
  int i = blockIdx.x * blockDim.x + threadIdx.x;
  if (i >= rows * kCm) return;
  int row = i / kCm;
  s[i] = s[i] / fmaxf(cnt[row], 1.f);
}

// out = mean @ Wl + bl + xd @ Wr ; mean flat rows, xd tensor (B,Cm,T1,N)
// grid.x = M tiles; block = 128 (4 waves = 4 col tiles of the 64 cols).
__global__ void sage_gemm_kernel(const float* __restrict__ mean,
                                 const float* __restrict__ xd, int N,
                                 const float* __restrict__ Wl,
                                 const float* __restrict__ bl,
                                 const float* __restrict__ Wr,
                                 float* __restrict__ out, int Mrows) {
  int lane = threadIdx.x & 31;
  int c0 = (threadIdx.x >> 5) * 16;
  int m0 = blockIdx.x * 16;
  int hf = lane >> 4, idx = lane & 15;
  int mA = m0 + idx;
  int mc = (mA < Mrows) ? mA : (Mrows - 1);
  int nn = mc % N;
  int tt = (mc / N) % kT1;
  int b = mc / (N * kT1);
  const size_t cstride = (size_t)kT1 * N;
  const float* pm = mean + (size_t)mc * kCm + 2 * hf;
  const float* px = xd + (((size_t)b * kCm) * kT1 + tt) * N + nn +
                    (size_t)(2 * hf) * cstride;
  const float* pl = Wl + (2 * hf) * kCm + c0 + idx;
  const float* pr = Wr + (2 * hf) * kCm + c0 + idx;

  v8f acc = {};
  for (int ks = 0; ks < kCm / 4; ++ks) {
    v2f a1, a2, b1, b2;
    a1[0] = pm[0];       a1[1] = pm[1];
    a2[0] = px[0];       a2[1] = px[cstride];
    b1[0] = pl[0];       b1[1] = pl[kCm];
    b2[0] = pr[0];       b2[1] = pr[kCm];
    acc = wmma4(a1, b1, acc);
    acc = wmma4(a2, b2, acc);
    pm += 4;
    px += 4 * cstride;
    pl += 4 * kCm;
    pr += 4 * kCm;
  }
#pragma unroll
  for (int r = 0; r < 8; ++r) {
    int m = m0 + r + 8 * hf;
    if (m >= Mrows) continue;
    int c = c0 + idx;
    out[(size_t)m * kCm + c] = acc[r] + bl[c];
  }
}

// out(M,64) = A(M,64 flat) @ W(64,64)
__global__ void gemm64_kernel(const float* __restrict__ A,
                              const float* __restrict__ W,
                              float* __restrict__ out, int Mrows) {
  int lane = threadIdx.x & 31;
  int c0 = (threadIdx.x >> 5) * 16;
  int m0 = blockIdx.x * 16;
  int hf = lane >> 4, idx = lane & 15;
  int mA = m0 + idx;
  int mc = (mA < Mrows) ? mA : (Mrows - 1);
  const float* pa = A + (size_t)mc * kCm + 2 * hf;
  const float* pw = W + (2 * hf) * kCm + c0 + idx;

  v8f acc = {};
  for (int ks = 0; ks < kCm / 4; ++ks) {
    v2f a, bv;
    a[0] = pa[0];  a[1] = pa[1];
    bv[0] = pw[0]; bv[1] = pw[kCm];
    acc = wmma4(a, bv, acc);
    pa += 4;
    pw += 4 * kCm;
  }
#pragma unroll
  for (int r = 0; r < 8; ++r) {
    int m = m0 + r + 8 * hf;
    if (m >= Mrows) continue;
    out[(size_t)m * kCm + c0 + idx] = acc[r];
  }
}

// small projection: out[(b*T1+t)*64+c] = sum_k x[b,k,t,0] * W[k][c] + bias[c]
__global__ void proj_kernel(const float* __restrict__ x,
                            const float* __restrict__ W,
                            const float* __restrict__ bias,
                            float* __restrict__ out) {
  int i = blockIdx.x * blockDim.x + threadIdx.x;
  if (i >= kB * kT1 * kCm) return;
  int c = i % kCm;
  int t = (i / kCm) % kT1;
  int b = i / (kCm * kT1);
  float acc = bias[c];
  for (int k = 0; k < kCm; ++k)
    acc += x[((size_t)b * kCm + k) * kT1 + t] * W[k * kCm + c];
  out[i] = acc;
}

__global__ void wm_scatter_kernel(const int* __restrict__ dstv,
                                  const float* __restrict__ w,
                                  float* __restrict__ Wm, int E) {
  int e = blockIdx.x * blockDim.x + threadIdx.x;
  if (e < E) atomicAdd(&Wm[dstv[e]], w[e]);
}

__global__ void room_combine_kernel(float* __restrict__ rflat,
                                    const float* __restrict__ tproj,
                                    const float* __restrict__ oproj,
                                    const float* __restrict__ Wm, int Mrows) {
  int i = blockIdx.x * blockDim.x + threadIdx.x;
  if (i >= Mrows * kCm) return;
  int c = i % kCm;
  int row = i / kCm;
  int nn = row % kNr;
  int t = (row / kNr) % kT1;
  int b = row / (kNr * kT1);
  int pti = (b * kT1 + t) * kCm + c;
  rflat[i] = rflat[i] + tproj[pti] + oproj[pti] * Wm[b * kNr + nn];
}

__global__ void deg_scatter_kernel(const int* __restrict__ ei1,
                                   const float* __restrict__ ew,
                                   float* __restrict__ deg, int E) {
  int gid = blockIdx.x * blockDim.x + threadIdx.x;
  if (gid >= kT1 * E) return;
  int e = gid % E;
  int t = gid / E;
  atomicAdd(&deg[ei1[e] + t * kB * kNr], ew[e]);
}

__global__ void gcn_self_kernel(const float* __restrict__ h,
                                const float* __restrict__ deg,
                                float* __restrict__ acc, int rows) {
  int i = blockIdx.x * blockDim.x + threadIdx.x;
  if (i >= rows * kCm) return;
  int row = i / kCm;
  acc[i] = h[i] / deg[row];  // self loop: dinv*1*dinv = 1/deg
}

__global__ void gcn_scatter_kernel(const float* __restrict__ h,
                                   const float* __restrict__ deg,
                                   const int* __restrict__ ei0,
                                   const int* __restrict__ ei1,
                                   const float* __restrict__ ew,
                                   float* __restrict__ acc, int E) {
  int gid = blockIdx.x * blockDim.x + threadIdx.x;
  int total = kT1 * E * 16;
  if (gid >= total) return;
  int sub = gid & 15;
  int ee = gid >> 4;
  int e = ee % E;
  int t = ee / E;
  int s = ei0[e] + t * kB * kNr;
  int d = ei1[e] + t * kB * kNr;
  float norm = rsqrtf(deg[s]) * ew[e] * rsqrtf(deg[d]);
#pragma unroll
  for (int j = 0; j < 4; ++j) {
    int c = sub * 4 + j;
    atomicAdd(&acc[(size_t)d * kCm + c], h[(size_t)s * kCm + c] * norm);
  }
}

__global__ void gcn_combine_kernel(float* __restrict__ rflat,
                                   const float* __restrict__ acc1,
                                   const float* __restrict__ acc2,
                                   const float* __restrict__ b1,
                                   const float* __restrict__ b2, int rows) {
  int i = blockIdx.x * blockDim.x + threadIdx.x;
  if (i >= rows * kCm) return;
  int c = i % kCm;
  rflat[i] = acc1[i] + b1[c] + acc2[i] + b2[c];
}

// ---------------- host orchestration ----------------------------------------
extern "C" void kernel_launch(void* const* d_in, const int* in_sizes, int n_in,
                              void* d_out, int out_size, void* d_ws,
                              size_t ws_size, hipStream_t stream) {
  (void)in_sizes; (void)n_in; (void)out_size; (void)ws_size;
  const float* x_room     = (const float*)d_in[0];
  const float* x_device   = (const float*)d_in[1];
  const float* x_property = (const float*)d_in[2];
  const float* x_outside  = (const float*)d_in[3];
  const float* x_time     = (const float*)d_in[4];
  const float* temp1_W = (const float*)d_in[5];
  const float* temp1_b = (const float*)d_in[6];
  const float* temp2_W = (const float*)d_in[7];
  const float* temp2_b = (const float*)d_in[8];
  const float* sage1_Wl = (const float*)d_in[9];
  const float* sage1_bl = (const float*)d_in[10];
  const float* sage1_Wr = (const float*)d_in[11];
  const float* sage2_Wl = (const float*)d_in[12];
  const float* sage2_bl = (const float*)d_in[13];
  const float* sage2_Wr = (const float*)d_in[14];
  const float* time_W = (const float*)d_in[15];
  const float* time_b = (const float*)d_in[16];
  const float* out_W = (const float*)d_in[17];
  const float* out_b = (const float*)d_in[18];
  const float* gcnh_W = (const float*)d_in[19];
  const float* gcnh_b = (const float*)d_in[20];
  const float* gcnv_W = (const float*)d_in[21];
  const float* gcnv_b = (const float*)d_in[22];
  const float* ln_g = (const float*)d_in[23];
  const float* ln_b = (const float*)d_in[24];
  const float* w_out = (const float*)d_in[25];
  const float* ew_rh = (const float*)d_in[26];
  const float* ew_rv = (const float*)d_in[27];
  const int* ei_p2d = (const int*)d_in[28];
  const int* ei_d2r = (const int*)d_in[29];
  const int* ei_out_dst = (const int*)d_in[30];
  const int* ei_rh = (const int*)d_in[31];
  const int* ei_rv = (const int*)d_in[32];

  // workspace layout (floats)
  constexpr size_t SZ_P1 = (size_t)kB * kCm * kT1 * kNp;
  constexpr size_t SZ_D1 = (size_t)kB * kCm * kT1 * kNd;
  constexpr size_t SZ_R1 = (size_t)kB * kCm * kT1 * kNr;
  constexpr size_t SZ_O1 = (size_t)kB * kCm * kT1;
  constexpr size_t SZ_S = (size_t)kB * kT1 * kNd * kCm;
  constexpr size_t SZ_CNT = (size_t)kB * kT1 * kNd;
  constexpr size_t SZ_DFLAT = SZ_D1;
  constexpr size_t SZ_RFLAT = SZ_R1;
  constexpr size_t SZ_PROJ = (size_t)kB * kT1 * kCm;
  constexpr size_t SZ_WM = (size_t)kB * kNr;
  constexpr size_t SZ_DEG = (size_t)kB * kT1 * kNr;

  float* ws = (float*)d_ws;
  size_t off = 0;
  float* p1 = ws + off; off += SZ_P1;
  float* d1 = ws + off; off += SZ_D1;
  float* r1 = ws + off; off += SZ_R1;
  float* o1 = ws + off; off += SZ_O1;
  float* t1 = ws + off; off += SZ_O1;
  float* aggS = ws + off; off += SZ_S;
  float* aggC = ws + off; off += SZ_CNT;
  float* dflat = ws + off; off += SZ_DFLAT;
  float* rflat = ws + off; off += SZ_RFLAT;
  float* tproj = ws + off; off += SZ_PROJ;
  float* oproj = ws + off; off += SZ_PROJ;
  float* Wm = ws + off; off += SZ_WM;
  float* deg = ws + off; off += SZ_DEG;
  // GCN scratch aliased onto d1 (dead after sage1 GEMM)
  float* hbuf = d1;
  float* acc1 = d1 + SZ_RFLAT;
  float* acc2 = d1 + 2 * SZ_RFLAT;

  auto fill = [&](float* p, float v, size_t n) {
    fill_kernel<<<dim3((unsigned)((n + 255) / 256)), 256, 0, stream>>>(p, v, n);
  };

  // ---- stage 1 temporal convs ----
  auto tc1 = [&](const float* x, int type, int N, float* out) {
    int M = kB * kT1 * N;
    int mt = (M + 15) / 16;
    tconv1_kernel<<<dim3(mt), 128, 0, stream>>>(
        x, temp1_W + (size_t)type * 128 * kCin * kKt, temp1_b + type * 128, out,
        N, M);
  };
  tc1(x_room, 0, kNr, r1);
  tc1(x_device, 1, kNd, d1);
  tc1(x_property, 2, kNp, p1);
  tc1(x_outside, 3, 1, o1);
  tc1(x_time, 4, 1, t1);

  // ---- SAGE 1: property -> device ----
  {
    int rows = kB * kT1 * kNd;
    fill(aggS, 0.f, (size_t)rows * kCm);
    fill(aggC, 0.f, rows);
    int tot = kT1 * 16384 * 16;
    sage_scatter_kernel<<<(tot + 255) / 256, 256, 0, stream>>>(
        p1, 0, ei_p2d, ei_p2d + 16384, 16384, kNp, kNd, aggS, aggC);
    mean_div_kernel<<<(rows * kCm + 255) / 256, 256, 0, stream>>>(aggS, aggC,
                                                                  rows);
    int mt = (rows + 15) / 16;
    sage_gemm_kernel<<<dim3(mt), 128, 0, stream>>>(
        aggS, d1, kNd, sage1_Wl, sage1_bl, sage1_Wr, dflat, rows);
  }
  // ---- SAGE 2: device -> room ----
  {
    int rows = kB * kT1 * kNr;
    fill(aggS, 0.f, (size_t)rows * kCm);
    fill(aggC, 0.f, rows);
    int tot = kT1 * 4096 * 16;
    sage_scatter_kernel<<<(tot + 255) / 256, 256, 0, stream>>>(
        dflat, 1, ei_d2r, ei_d2r + 4096, 4096, kNd, kNr, aggS, aggC);
    mean_div_kernel<<<(rows * kCm + 255) / 256, 256, 0, stream>>>(aggS, aggC,
                                                                  rows);
    int mt = (rows + 15) / 16;
    sage_gemm_kernel<<<dim3(mt), 128, 0, stream>>>(
        aggS, r1, kNr, sage2_Wl, sage2_bl, sage2_Wr, rflat, rows);
  }

  // ---- time/outside projections + Wm scatter + combine ----
  {
    int tot = kB * kT1 * kCm;
    proj_kernel<<<(tot + 255) / 256, 256, 0, stream>>>(t1, time_W, time_b,
                                                       tproj);
    proj_kernel<<<(tot + 255) / 256, 256, 0, stream>>>(o1, out_W, out_b, oproj);
  }
  fill(Wm, 0.f, SZ_WM);
  wm_scatter_kernel<<<(1024 + 255) / 256, 256, 0, stream>>>(ei_out_dst, w_out,
                                                            Wm, 1024);
  {
    int rows = kB * kT1 * kNr;
    room_combine_kernel<<<(rows * kCm + 255) / 256, 256, 0, stream>>>(
        rflat, tproj, oproj, Wm, rows);
  }

  // ---- dual GCN on rooms ----
  auto gcn = [&](const float* Wg, const int* ei, const float* ew, int E,
                 float* acc) {
    int rows = kB * kT1 * kNr;
    fill(deg, 1.0f, rows);  // self loops
    deg_scatter_kernel<<<(kT1 * E + 255) / 256, 256, 0, stream>>>(ei + E, ew,
                                                                  deg, E);
    int mt = (rows + 15) / 16;
    gemm64_kernel<<<dim3(mt), 128, 0, stream>>>(rflat, Wg, hbuf, rows);
    gcn_self_kernel<<<(rows * kCm + 255) / 256, 256, 0, stream>>>(hbuf, deg,
                                                                  acc, rows);
    int tot = kT1 * E * 16;
    gcn_scatter_kernel<<<(tot + 255) / 256, 256, 0, stream>>>(
        hbuf, deg, ei, ei + E, ew, acc, E);
  };
  gcn(gcnh_W, ei_rh, ew_rh, 1024, acc1);
  gcn(gcnv_W, ei_rv, ew_rv, 512, acc2);
  {
    int rows = kB * kT1 * kNr;
    gcn_combine_kernel<<<(rows * kCm + 255) / 256, 256, 0, stream>>>(
        rflat, acc1, acc2, gcnh_b, gcnv_b, rows);
  }

  // ---- stage 2 temporal convs (fused relu) + LayerNorm into d_out ----
  size_t obase = 0;
  auto tc2 = [&](const float* x, int layout, int type, int N) {
    int M = kB * kT2 * N;
    int mt = (M + 15) / 16;
    float* outp = (float*)d_out + obase;
    const float* Wp = temp2_W + (size_t)type * 128 * kCm * kKt;
    const float* bp = temp2_b + type * 128;
    if (layout == 0)
      tconv2_kernel<0><<<dim3(mt), 128, 0, stream>>>(x, Wp, bp, outp, N, M);
    else
      tconv2_kernel<1><<<dim3(mt), 128, 0, stream>>>(x, Wp, bp, outp, N, M);
    ln_kernel<<<(M + 255) / 256, 256, 0, stream>>>(outp, ln_g + type * kCo,
                                                   ln_b + type * kCo, N, M);
    obase += (size_t)kB * kCo * kT2 * N;
  };
  tc2(rflat, 1, 0, kNr);
  tc2(dflat, 1, 1, kNd);
  tc2(p1, 0, 2, kNp);
  tc2(o1, 0, 3, 1);
  tc2(t1, 0, 4, 1);
}